// PathGuidedAggregator_52682068853458
// MI455X (gfx1250) — compile-verified
//
#include <hip/hip_runtime.h>
#include <cstdint>

// Problem constants (from the reference)
#define DIM   128   // embed dim D
#define PATHS 16    // P
#define KMAX  16    // K
#define HID   64    // H = D/2
#define W1LD  72    // LDS row stride for W1 (2*72 mod 64 = 16 -> conflict-free B frags)

typedef float v2f __attribute__((ext_vector_type(2)));
typedef float v4f __attribute__((ext_vector_type(4)));   // native vector: OK for nontemporal builtins
typedef float v8f __attribute__((ext_vector_type(8)));

// ---------------------------------------------------------------------------
// Kernel 1: zero the [N, D] output with non-temporal stores (keep the
// embedding table resident in the 192MB L2; the output is stream-once).
// ---------------------------------------------------------------------------
__global__ void zero_out_kernel(v4f* __restrict__ out, long long n4) {
  long long i = (long long)blockIdx.x * blockDim.x + threadIdx.x;
  const long long stride = (long long)gridDim.x * blockDim.x;
  const v4f z = {0.f, 0.f, 0.f, 0.f};
  for (; i < n4; i += stride) __builtin_nontemporal_store(z, &out[i]);
}

// ---------------------------------------------------------------------------
// Kernel 2: one workgroup (8 wave32) per sparse node.
//   Phase 0: stage W1 into LDS (coalesced float4, stride-72 rows)
//   Phase 1: masked-mean gather of endpoint embeddings  -> s_agg[16][128]
//   Phase 2: [16,128] @ [128,64] via V_WMMA_F32_16X16X4_F32 (A and B from LDS)
//   Phase 3: h @ W2 + b2, sigmoid, validity masking, 1/max(nv,1)
//   Phase 4: weighted mean over paths, non-temporal scatter to out[sid[s]]
// ---------------------------------------------------------------------------
__global__ __launch_bounds__(256) void path_node_kernel(
    const float* __restrict__ E,      // [N, 128] entity embeds (L2-resident)
    const float* __restrict__ W1,     // [128, 64] row-major
    const float* __restrict__ b1,     // [64]
    const float* __restrict__ W2,     // [64, 1]
    const float* __restrict__ b2,     // [1]
    const int*   __restrict__ sid,    // [S]
    const int*   __restrict__ eids,   // [S, 16, 16]
    const unsigned char* __restrict__ emask,  // [S, 16, 16] (jax bool = 1 byte)
    float* __restrict__ out)          // [N, 128]
{
  __shared__ float s_W1[DIM * W1LD];        // 36.9 KB, bank-conflict-free frags
  __shared__ float s_agg[PATHS][DIM + 4];   // +4 pad: A-frag lanes hit distinct banks
  __shared__ float s_h[PATHS][HID + 4];
  __shared__ float s_w[PATHS];
  __shared__ int   s_cnt[PATHS];
  __shared__ float s_invnv;

  const int s    = blockIdx.x;
  const int tid  = threadIdx.x;
  const int lane = tid & 31;
  const int wv   = tid >> 5;    // wave id 0..7 (wave32)

  // ---- Phase 0: stage W1 -> LDS (issued first so loads overlap the gather).
  // 8192 floats = 2048 float4; 256 threads x 8 iters; rows are 64 floats so
  // each float4 stays inside one row.
  #pragma unroll
  for (int it = 0; it < 8; ++it) {
    const int i4 = it * 256 + tid;            // float4 index 0..2047
    const int r  = i4 >> 4;                   // row = (i4*4)/64
    const int c  = (i4 & 15) << 2;            // col = (i4*4)%64
    const v4f v = *(const v4f*)(W1 + i4 * 4);
    *(v4f*)(&s_W1[r * W1LD + c]) = v;
  }

  // ---- Phase 1: gather + masked mean. Wave wv handles paths wv and wv+8.
  #pragma unroll
  for (int pp = 0; pp < 2; ++pp) {
    const int p    = wv + pp * 8;
    const int base = (s * PATHS + p) * KMAX;
    v4f acc = {0.f, 0.f, 0.f, 0.f};
    int cnt = 0;
    for (int k = 0; k < KMAX; ++k) {
      // index/mask are stream-once: non-temporal, keep L2 for the table
      const unsigned char mk = __builtin_nontemporal_load(&emask[base + k]);
      if (mk) {                                    // wave-uniform branch
        const int id = __builtin_nontemporal_load(&eids[base + k]);
        const v4f v = *(const v4f*)(E + (long long)id * DIM + lane * 4);
        acc += v;
        ++cnt;
      }
    }
    const float inv = 1.0f / (float)max(cnt, 1);
    s_agg[p][lane * 4 + 0] = acc.x * inv;
    s_agg[p][lane * 4 + 1] = acc.y * inv;
    s_agg[p][lane * 4 + 2] = acc.z * inv;
    s_agg[p][lane * 4 + 3] = acc.w * inv;
    if (lane == 0) s_cnt[p] = cnt;
  }
  __syncthreads();

  // ---- Phase 2: h = relu(agg @ W1 + b1) with f32 WMMA. Waves 0..3, one
  // 16-wide N-tile each, K swept in 32 steps of 4. A and B both from LDS.
  if (wv < 4) {                        // uniform per wave -> EXEC all-1s inside
    const int nt    = wv;
    const int row   = lane & 15;              // A-row / B-col within tile
    const int khalf = (lane >> 4) << 1;       // lanes 16-31 hold K+2, K+3
    const int col   = nt * 16 + row;
    v8f c = {};
    for (int kk = 0; kk < DIM; kk += 4) {
      v2f a, b;
      // A 16x4 f32 layout: VGPR0 = K=0 (lanes 0-15) / K=2 (16-31), VGPR1 = K=1/K=3
      a.x = s_agg[row][kk + khalf];
      a.y = s_agg[row][kk + khalf + 1];
      // B 4x16 f32: same K striping, N across lanes; from LDS stride-72 rows
      b.x = s_W1[(kk + khalf) * W1LD + col];
      b.y = s_W1[(kk + khalf + 1) * W1LD + col];
      c = __builtin_amdgcn_wmma_f32_16x16x4_f32(
          /*neg_a=*/false, a, /*neg_b=*/false, b,
          /*c_mod=*/(short)0, c, /*reuse_a=*/false, /*reuse_b=*/false);
    }
    const float bias = b1[col];
    #pragma unroll
    for (int r = 0; r < 8; ++r) {
      // C/D f32 16x16 layout: VGPR r -> M=r (lanes 0-15) / M=r+8 (lanes 16-31)
      const int m = r + ((lane >> 4) << 3);
      const float hv = c[r] + bias;
      s_h[m][col] = hv > 0.f ? hv : 0.f;
    }
  }
  __syncthreads();

  // ---- Phase 3: per-path scalar weight = sigmoid(h . W2 + b2), masked by cnt>0.
  if (tid < PATHS) {
    float sum = 0.f;
    #pragma unroll 8
    for (int n = 0; n < HID; ++n) sum += s_h[tid][n] * W2[n];
    sum += b2[0];
    const float wgt = 1.0f / (1.0f + __expf(-sum));
    s_w[tid] = (s_cnt[tid] > 0) ? wgt : 0.f;
  }
  __syncthreads();
  if (tid == 0) {
    int nv = 0;
    #pragma unroll
    for (int m = 0; m < PATHS; ++m) nv += (s_cnt[m] > 0) ? 1 : 0;
    s_invnv = 1.0f / (float)max(nv, 1);   // nv==0 -> all s_w are 0 anyway
  }
  __syncthreads();

  // ---- Phase 4: node_feat = (sum_p w_p * agg_p) / max(nv,1); scatter row.
  if (tid < DIM) {
    float acc = 0.f;
    #pragma unroll
    for (int m = 0; m < PATHS; ++m) acc += s_w[m] * s_agg[m][tid];
    acc *= s_invnv;
    __builtin_nontemporal_store(acc, &out[(long long)sid[s] * DIM + tid]);
  }
}

// ---------------------------------------------------------------------------
extern "C" void kernel_launch(void* const* d_in, const int* in_sizes, int n_in,
                              void* d_out, int out_size, void* d_ws, size_t ws_size,
                              hipStream_t stream) {
  const float* E   = (const float*)d_in[0];  // entity_embeds [N,128]
  const float* W1  = (const float*)d_in[1];  // [128,64]
  const float* b1  = (const float*)d_in[2];  // [64]
  const float* W2  = (const float*)d_in[3];  // [64,1]
  const float* b2  = (const float*)d_in[4];  // [1]
  const int*   sid = (const int*)d_in[5];    // [S]
  const int*   eid = (const int*)d_in[6];    // [S,16,16]
  const unsigned char* msk = (const unsigned char*)d_in[7];  // [S,16,16] bool
  float* out = (float*)d_out;

  const int S = in_sizes[5];                 // 8192
  const long long n4 = (long long)out_size / 4;

  zero_out_kernel<<<2048, 256, 0, stream>>>((v4f*)out, n4);
  path_node_kernel<<<S, 256, 0, stream>>>(E, W1, b1, W2, b2, sid, eid, msk, out);
}